// AtomConvLayer_19078244729258
// MI455X (gfx1250) — compile-verified
//
#include <hip/hip_runtime.h>
#include <hip/hip_bf16.h>

typedef float v2f __attribute__((ext_vector_type(2)));
typedef float v8f __attribute__((ext_vector_type(8)));

#define NATOM   100000
#define MNBR    12
#define FDIM    64
#define KDIM    192          // 2F + NBR
#define ODIM    128          // 2F
#define ROWS    (NATOM * MNBR)          // 1,200,000
#define ROWS_PER_BLOCK 96               // 8 atoms * 12 nbrs
#define NBLK    (ROWS / ROWS_PER_BLOCK) // 12,500
#define WAVES   6
#define THREADS (WAVES * 32)
#define NCHUNK  (KDIM / 4)              // 48 K-chunks of 4

__device__ __forceinline__ float softplus_f(float x) {
    float ax = fabsf(x);
    return fmaxf(x, 0.0f) + log1pf(expf(-ax));
}
__device__ __forceinline__ float sigmoid_f(float x) {
    return 1.0f / (1.0f + expf(-x));
}

// ---------------------------------------------------------------------------
// B-fragment swizzle: W (128x192, row-major) -> LDS pairs laid out
//   [chunk ck 0..47][phase p 0..1][nt 0..7][lane16 0..15] : (W[n][k], W[n][k+1])
// with n = nt*16+lane16, k = ck*4 + 2p.  Inside the GEMM loop each lane's
// B-fragment address is (one base VGPR + compile-time immediate).
// ---------------------------------------------------------------------------
__device__ __forceinline__ void load_W_swizzled(const float* __restrict__ Wg,
                                                float* __restrict__ Blds) {
    for (int i = threadIdx.x; i < NCHUNK * 2 * 8 * 16; i += blockDim.x) {
        const int l16 = i & 15;
        const int nt  = (i >> 4) & 7;
        const int p   = (i >> 7) & 1;
        const int ck  = i >> 8;
        const int n = nt * 16 + l16;
        const int k = ck * 4 + p * 2;
        *(v2f*)(Blds + (size_t)i * 2) = *(const v2f*)(Wg + n * KDIM + k);
    }
}

// ---------------------------------------------------------------------------
// One wave computes a 16-row x 128-col tile of gated = total @ W^T using
// V_WMMA_F32_16X16X4_F32, K = 192 split into 3 fixed-source regions.
// A-frag (32-bit A 16x4): lane(l&15)=row, lanes>=16 hold K=kk+2,kk+3.
// C/D: VGPR j -> row = j + (l>=16 ? 8:0), col = l&15.
// ---------------------------------------------------------------------------
__device__ __forceinline__ void gemm_tile(const float* __restrict__ atom_fea,
                                          const float* __restrict__ nbr_fea,
                                          const int*   __restrict__ nidx,
                                          const float* __restrict__ Blds,
                                          int tile_row0, int lane, v8f acc[8]) {
    const int mr   = lane & 15;
    const int p    = lane >> 4;          // phase: 0 or 1
    const int koff = p * 2;
    const int ra   = tile_row0 + mr;
    const int ia   = ra / MNBR;
    const int ga   = nidx[ra];
    const float* pSelf = atom_fea + ia * FDIM + koff;
    const float* pGath = atom_fea + ga * FDIM + koff;
    const float* pNbr  = nbr_fea  + (size_t)ra * FDIM + koff;

    // per-(ck,p) group = 256 floats; per-ck (both phases) = 512 floats
    const float* bb = Blds + p * 256 + mr * 2;
    const float* b0 = bb;                 // region 0: chunks  0..15
    const float* b1 = bb + 16 * 512;      // region 1: chunks 16..31
    const float* b2 = bb + 32 * 512;      // region 2: chunks 32..47

    #pragma unroll
    for (int ck = 0; ck < 16; ++ck) {
        const v2f a = *(const v2f*)(pSelf + ck * 4);
        #pragma unroll
        for (int nt = 0; nt < 8; ++nt) {
            const v2f b = *(const v2f*)(b0 + ck * 512 + nt * 32);
            acc[nt] = __builtin_amdgcn_wmma_f32_16x16x4_f32(
                false, a, false, b, (short)0, acc[nt], false, false);
        }
    }
    #pragma unroll
    for (int ck = 0; ck < 16; ++ck) {
        const v2f a = *(const v2f*)(pGath + ck * 4);
        #pragma unroll
        for (int nt = 0; nt < 8; ++nt) {
            const v2f b = *(const v2f*)(b1 + ck * 512 + nt * 32);
            acc[nt] = __builtin_amdgcn_wmma_f32_16x16x4_f32(
                false, a, false, b, (short)0, acc[nt], false, false);
        }
    }
    #pragma unroll
    for (int ck = 0; ck < 16; ++ck) {
        const v2f a = *(const v2f*)(pNbr + ck * 4);
        #pragma unroll
        for (int nt = 0; nt < 8; ++nt) {
            const v2f b = *(const v2f*)(b2 + ck * 512 + nt * 32);
            acc[nt] = __builtin_amdgcn_wmma_f32_16x16x4_f32(
                false, a, false, b, (short)0, acc[nt], false, false);
        }
    }
}

// ---------------------------------------------------------------------------
// Pass 1: GEMM + deterministic per-channel sum / sumsq partials (incl. bias)
// P1[block][0..127]=sum, [128..255]=sumsq
// ---------------------------------------------------------------------------
__global__ void __launch_bounds__(THREADS)
pass1_stats(const float* __restrict__ atom_fea, const float* __restrict__ nbr_fea,
            const int* __restrict__ nidx, const float* __restrict__ Wg,
            const float* __restrict__ bias, float* __restrict__ P1) {
    extern __shared__ float smem[];
    float* Blds = smem;                       // 24576 floats
    float* Sl   = smem + ODIM * KDIM;         // [6][32][16] = 3072 floats

    load_W_swizzled(Wg, Blds);
    __syncthreads();

    const int wv   = threadIdx.x >> 5;
    const int lane = threadIdx.x & 31;
    const int tile_row0 = blockIdx.x * ROWS_PER_BLOCK + wv * 16;

    v8f acc[8] = {};
    gemm_tile(atom_fea, nbr_fea, nidx, Blds, tile_row0, lane, acc);

    #pragma unroll
    for (int nt = 0; nt < 8; ++nt) {
        const int c  = nt * 16 + (lane & 15);
        const float bc = bias[c];
        float s = 0.0f, q = 0.0f;
        #pragma unroll
        for (int j = 0; j < 8; ++j) {
            const float g = acc[nt][j] + bc;
            s += g; q += g * g;
        }
        Sl[(wv * 32 + lane) * 16 + nt * 2 + 0] = s;
        Sl[(wv * 32 + lane) * 16 + nt * 2 + 1] = q;
    }
    __syncthreads();

    // fixed-order combine -> deterministic
    for (int ch = threadIdx.x; ch < 2 * ODIM; ch += THREADS) {
        const int stat = ch >> 7;
        const int c    = ch & 127;
        const int nt   = c >> 4;
        const int li   = c & 15;
        float a = 0.0f;
        #pragma unroll
        for (int w2 = 0; w2 < WAVES; ++w2) {
            a += Sl[(w2 * 32 + li)      * 16 + nt * 2 + stat];
            a += Sl[(w2 * 32 + li + 16) * 16 + nt * 2 + stat];
        }
        P1[(size_t)blockIdx.x * 256 + ch] = a;
    }
}

// ---------------------------------------------------------------------------
__global__ void reduce_partials(const float* __restrict__ src, float* __restrict__ dst,
                                int nrows, int nchan2) {
    const int c = threadIdx.x;
    const int b = blockIdx.x;
    float s = 0.0f;
    for (int r = b; r < nrows; r += gridDim.x)
        s += src[(size_t)r * nchan2 + c];
    dst[(size_t)b * nchan2 + c] = s;
}

__global__ void finalize_bn(const float* __restrict__ part, int nparts, int nchan,
                            float invR, const float* __restrict__ gamma,
                            const float* __restrict__ beta, float* __restrict__ ss) {
    const int c = threadIdx.x;
    if (c >= nchan) return;
    float s = 0.0f, q = 0.0f;
    for (int r = 0; r < nparts; ++r) {
        s += part[(size_t)r * 2 * nchan + c];
        q += part[(size_t)r * 2 * nchan + nchan + c];
    }
    const float mean = s * invR;
    float var = q * invR - mean * mean;
    var = fmaxf(var, 0.0f);
    const float sc = gamma[c] * rsqrtf(var + 1e-5f);
    ss[c]         = sc;
    ss[nchan + c] = beta[c] - mean * sc;
}

// ---------------------------------------------------------------------------
// Pass 2: recompute GEMM, apply BN1 + sigmoid/softplus * bond_weights,
// deterministic neighbor-sum (8 atoms/block) -> NS, plus BN2 partials P2.
// ---------------------------------------------------------------------------
__global__ void __launch_bounds__(THREADS)
pass2_fused(const float* __restrict__ atom_fea, const float* __restrict__ nbr_fea,
            const int* __restrict__ nidx, const float* __restrict__ Wg,
            const float* __restrict__ bias, const float* __restrict__ bondw,
            const float* __restrict__ ss1, float* __restrict__ NS,
            float* __restrict__ P2) {
    extern __shared__ float smem[];
    float* Blds = smem;                              // 24576 floats
    float* cl   = smem + ODIM * KDIM;                // [96][65] = 6240 floats
    float* nsl  = cl + ROWS_PER_BLOCK * 65;          // [8][64]  = 512 floats

    load_W_swizzled(Wg, Blds);
    __syncthreads();

    const int wv   = threadIdx.x >> 5;
    const int lane = threadIdx.x & 31;
    const int hi   = lane >> 4;
    const int tile_row0 = blockIdx.x * ROWS_PER_BLOCK + wv * 16;

    v8f acc[8] = {};
    gemm_tile(atom_fea, nbr_fea, nidx, Blds, tile_row0, lane, acc);

    float bwj[8];
    #pragma unroll
    for (int j = 0; j < 8; ++j)
        bwj[j] = bondw[tile_row0 + hi * 8 + j];

    #pragma unroll
    for (int nt = 0; nt < 4; ++nt) {
        const int cf = nt * 16 + (lane & 15);
        const int cc = cf + FDIM;
        const float scF = ss1[cf];
        const float shF = ss1[ODIM + cf] + bias[cf] * scF;
        const float scC = ss1[cc];
        const float shC = ss1[ODIM + cc] + bias[cc] * scC;
        #pragma unroll
        for (int j = 0; j < 8; ++j) {
            const float y1 = acc[nt][j]     * scF + shF;
            const float y2 = acc[nt + 4][j] * scC + shC;
            const float bw = bwj[j];
            const float v  = sigmoid_f(y1) * softplus_f(y2) * bw * bw;
            cl[(wv * 16 + hi * 8 + j) * 65 + cf] = v;
        }
    }
    __syncthreads();

    // deterministic neighbor-sum: 8 atoms x 64 channels
    for (int o = threadIdx.x; o < 8 * FDIM; o += THREADS) {
        const int a = o >> 6;
        const int c = o & 63;
        float s = 0.0f;
        #pragma unroll
        for (int m = 0; m < MNBR; ++m)
            s += cl[(a * MNBR + m) * 65 + c];
        nsl[a * FDIM + c] = s;
        NS[((size_t)blockIdx.x * 8 + a) * FDIM + c] = s;
    }
    __syncthreads();

    // BN2 partials over this block's 8 atom rows (fixed order)
    if (threadIdx.x < 2 * FDIM) {
        const int stat = threadIdx.x >> 6;
        const int c    = threadIdx.x & 63;
        float s = 0.0f;
        #pragma unroll
        for (int a = 0; a < 8; ++a) {
            const float v = nsl[a * FDIM + c];
            s += stat ? v * v : v;
        }
        P2[(size_t)blockIdx.x * 128 + threadIdx.x] = s;
    }
}

// ---------------------------------------------------------------------------
__global__ void final_out(const float* __restrict__ atom_fea,
                          const float* __restrict__ NS,
                          const float* __restrict__ ss2,
                          float* __restrict__ out, int total) {
    const int i = blockIdx.x * blockDim.x + threadIdx.x;
    if (i >= total) return;
    const int c = i & 63;
    const float v = atom_fea[i] + NS[i] * ss2[c] + ss2[FDIM + c];
    out[i] = softplus_f(v);
}

// ---------------------------------------------------------------------------
extern "C" void kernel_launch(void* const* d_in, const int* in_sizes, int n_in,
                              void* d_out, int out_size, void* d_ws, size_t ws_size,
                              hipStream_t stream) {
    const float* atom_fea = (const float*)d_in[0];
    const float* nbr_fea  = (const float*)d_in[1];
    const float* bondw    = (const float*)d_in[2];
    const int*   nidx     = (const int*)  d_in[3];
    const float* Wg       = (const float*)d_in[4];
    const float* bias     = (const float*)d_in[5];
    const float* gamma1   = (const float*)d_in[6];
    const float* beta1    = (const float*)d_in[7];
    const float* gamma2   = (const float*)d_in[8];
    const float* beta2    = (const float*)d_in[9];
    float*       out      = (float*)d_out;

    float* wsf = (float*)d_ws;
    float* P1  = wsf;                          // [12500][256]
    float* T1  = P1 + (size_t)NBLK * 256;      // [64][256]
    float* SS1 = T1 + 64 * 256;                // [256]
    float* NS  = SS1 + 256;                    // [N][64]
    float* P2  = NS + (size_t)NATOM * FDIM;    // [12500][128]
    float* T2  = P2 + (size_t)NBLK * 128;      // [64][128]
    float* SS2 = T2 + 64 * 128;                // [128]

    const size_t sh1 = (ODIM * KDIM + WAVES * 32 * 16) * sizeof(float);
    const size_t sh2 = (ODIM * KDIM + ROWS_PER_BLOCK * 65 + 8 * FDIM) * sizeof(float);

    pass1_stats<<<NBLK, THREADS, sh1, stream>>>(atom_fea, nbr_fea, nidx, Wg, bias, P1);
    reduce_partials<<<64, 256, 0, stream>>>(P1, T1, NBLK, 256);
    finalize_bn<<<1, 128, 0, stream>>>(T1, 64, ODIM, 1.0f / (float)ROWS, gamma1, beta1, SS1);

    pass2_fused<<<NBLK, THREADS, sh2, stream>>>(atom_fea, nbr_fea, nidx, Wg, bias,
                                                bondw, SS1, NS, P2);
    reduce_partials<<<64, 128, 0, stream>>>(P2, T2, NBLK, 128);
    finalize_bn<<<1, 64, 0, stream>>>(T2, 64, FDIM, 1.0f / (float)NATOM, gamma2, beta2, SS2);

    const int total = NATOM * FDIM;
    final_out<<<(total + 255) / 256, 256, 0, stream>>>(atom_fea, NS, SS2, out, total);
}